// GCNModel_2628519985884
// MI455X (gfx1250) — compile-verified
//
#include <hip/hip_runtime.h>
#include <math.h>

typedef float v2f __attribute__((ext_vector_type(2)));
typedef float v8f __attribute__((ext_vector_type(8)));

#define NN 100000   // nodes
#define NE 800000   // edges (w/o self loops)
#define NG 512      // graphs
#define DIN 128
#define DH  256
#define DMLP 512

// ---------------------------------------------------------------------------
// degree count: deg[dst[e]] += 1 (self-loop +1 folded into finalize)
// ---------------------------------------------------------------------------
__global__ void deg_count_kernel(const int* __restrict__ dst, float* __restrict__ deg, int E) {
    int e = blockIdx.x * blockDim.x + threadIdx.x;
    if (e < E) unsafeAtomicAdd(&deg[dst[e]], 1.0f);
}

__global__ void deg_finalize_kernel(float* __restrict__ isq, int n) {
    int i = blockIdx.x * blockDim.x + threadIdx.x;
    if (i < n) {
        float d = isq[i] + 1.0f;   // + self loop; always >= 1
        isq[i] = rsqrtf(d);
    }
}

// ---------------------------------------------------------------------------
// fp32 WMMA GEMM: C[M,Nn] = A[M,K] @ B[K,Nn]  (+bias, +relu optional)
// One wave computes a 32x64 macro-tile = 2x4 grid of 16x16 WMMA tiles.
// K, Nn compile-time -> immediate-offset loads; per K-step(4):
//   2x global_load_b64 (A frags) + 8x global_load_b32 (B frags) -> 8 wmma.
// Fragment layouts (V_WMMA_F32_16X16X4_F32, wave32):
//   A 16x4 : lane L -> m=L&15, kpair=(L>>4)*2, holds K=kp,kp+1
//   B 4x16 : lane L -> n=L&15, kpair=(L>>4)*2, holds K=kp,kp+1
//   C      : VGPR r -> row = r + 8*(L>>4), col = L&15
// ---------------------------------------------------------------------------
template<int K, int Nn, bool BIAS, bool RELU>
__global__ void wmma_gemm_f32_kernel(const float* __restrict__ A,
                                     const float* __restrict__ B,
                                     const float* __restrict__ bias,
                                     float* __restrict__ C, int M) {
    constexpr int WM = 32, WN = 64;
    const int lane = threadIdx.x & 31;
    const int wave = threadIdx.x >> 5;
    const int tilesN = Nn / WN;
    const int totalTiles = (M / WM) * tilesN;
    int tile = blockIdx.x * (blockDim.x >> 5) + wave;
    if (tile >= totalTiles) return;              // uniform per wave -> EXEC stays all-1s

    const int mt = tile / tilesN;
    const int nt = tile - mt * tilesN;

    const int l15 = lane & 15;
    const int hi  = lane >> 4;                   // 0/1
    const int kb  = hi << 1;                     // 0/2

    const float* __restrict__ aPtr0 = A + (size_t)(mt * WM + l15) * K + kb;
    const float* __restrict__ aPtr1 = aPtr0 + (size_t)16 * K;
    const float* __restrict__ bPtr  = B + (size_t)kb * Nn + nt * WN + l15;

    v8f acc[8] = {};                             // [mi*4 + nj]

    #pragma unroll 2
    for (int k0 = 0; k0 < K; k0 += 4) {
        v2f a0 = *(const v2f*)aPtr0;             // K=kb..kb+1, rows mt*32+l15
        v2f a1 = *(const v2f*)aPtr1;             // rows mt*32+16+l15
        #pragma unroll
        for (int nj = 0; nj < 4; ++nj) {
            v2f b;
            b.x = bPtr[nj * 16];                 // row k0+kb
            b.y = bPtr[nj * 16 + Nn];            // row k0+kb+1
            acc[nj] = __builtin_amdgcn_wmma_f32_16x16x4_f32(
                false, a0, false, b, (short)0, acc[nj], false, false);
            acc[4 + nj] = __builtin_amdgcn_wmma_f32_16x16x4_f32(
                false, a1, false, b, (short)0, acc[4 + nj], false, false);
        }
        aPtr0 += 4;
        aPtr1 += 4;
        bPtr  += (size_t)4 * Nn;
    }

    const int rowBase = mt * WM + (hi << 3);     // + r gives C row of acc element r
    const int colBase = nt * WN + l15;
    #pragma unroll
    for (int nj = 0; nj < 4; ++nj) {
        float bv;
        if constexpr (BIAS) bv = bias[colBase + nj * 16];
        #pragma unroll
        for (int mi = 0; mi < 2; ++mi) {
            float* __restrict__ Cp = C + (size_t)(rowBase + mi * 16) * Nn + colBase + nj * 16;
            v8f av = acc[mi * 4 + nj];
            #pragma unroll
            for (int r = 0; r < 8; ++r) {
                float v = av[r];
                if constexpr (BIAS) v += bv;
                if constexpr (RELU) v = fmaxf(v, 0.0f);
                Cp[(size_t)r * Nn] = v;
            }
        }
    }
}

// ---------------------------------------------------------------------------
// edge scatter: agg[dst] += h[src] * (isq[src]*isq[dst]); blocks E..E+N-1 are
// the self loops. One block per edge: 64 lanes x float4 = 256 channels.
// ---------------------------------------------------------------------------
__global__ void gcn_scatter_kernel(const int* __restrict__ src,
                                   const int* __restrict__ dst,
                                   const float* __restrict__ isq,
                                   const float* __restrict__ h,
                                   float* __restrict__ agg, int E) {
    const int e = blockIdx.x;
    int s, t;
    if (e < E) { s = src[e]; t = dst[e]; }
    else       { s = t = e - E; }
    const float nrm = isq[s] * isq[t];
    const float4* __restrict__ hs = (const float4*)(h + (size_t)s * DH);
    float* __restrict__       at  = agg + (size_t)t * DH;
    const int d = threadIdx.x;                   // 0..63 -> float4 index
    float4 v = hs[d];
    unsafeAtomicAdd(&at[4 * d + 0], v.x * nrm);
    unsafeAtomicAdd(&at[4 * d + 1], v.y * nrm);
    unsafeAtomicAdd(&at[4 * d + 2], v.z * nrm);
    unsafeAtomicAdd(&at[4 * d + 3], v.w * nrm);
}

// in-place h = relu(h + bias[col])   (D power of two)
__global__ void bias_relu_kernel(float* __restrict__ h, const float* __restrict__ bias,
                                 int total, int Dmask) {
    int i = blockIdx.x * blockDim.x + threadIdx.x;
    if (i < total) h[i] = fmaxf(h[i] + bias[i & Dmask], 0.0f);
}

// ---------------------------------------------------------------------------
// pooling: sums[batch[i]] += h[i]; cnt[batch[i]] += 1
// ---------------------------------------------------------------------------
__global__ void pool_accum_kernel(const float* __restrict__ h,
                                  const int* __restrict__ batch,
                                  float* __restrict__ sums,
                                  float* __restrict__ cnt) {
    const int i = blockIdx.x;
    const int g = batch[i];
    const float* __restrict__ hi = h    + (size_t)i * DH;
    float* __restrict__       sg = sums + (size_t)g * DH;
    for (int d = threadIdx.x; d < DH; d += blockDim.x)
        unsafeAtomicAdd(&sg[d], hi[d]);
    if (threadIdx.x == 0) unsafeAtomicAdd(&cnt[g], 1.0f);
}

// pooled = sums/max(cnt,1) + (cnt>0 ? b2 : 0)    (b2 folded: mean(x+b)=mean(x)+b)
__global__ void pool_final_kernel(float* __restrict__ sums, const float* __restrict__ cnt,
                                  const float* __restrict__ b2, int total, int Dmask, int Dshift) {
    int i = blockIdx.x * blockDim.x + threadIdx.x;
    if (i < total) {
        float c = cnt[i >> Dshift];
        float v = sums[i] / fmaxf(c, 1.0f);
        if (c > 0.0f) v += b2[i & Dmask];
        sums[i] = v;
    }
}

// ---------------------------------------------------------------------------
// head: out[g] = sigmoid(dot(z[g,:], w) + b)   (one block per graph)
// ---------------------------------------------------------------------------
__global__ void head_kernel(const float* __restrict__ z, const float* __restrict__ w,
                            const float* __restrict__ b, float* __restrict__ out, int Dm) {
    __shared__ float red[8];
    const int g = blockIdx.x;
    float acc = 0.0f;
    for (int d = threadIdx.x; d < Dm; d += blockDim.x)
        acc += z[(size_t)g * Dm + d] * w[d];
    #pragma unroll
    for (int off = 16; off > 0; off >>= 1)
        acc += __shfl_down(acc, off, 32);
    if ((threadIdx.x & 31) == 0) red[threadIdx.x >> 5] = acc;
    __syncthreads();
    if (threadIdx.x == 0) {
        float s = 0.0f;
        const int nw = blockDim.x >> 5;
        for (int wv = 0; wv < nw; ++wv) s += red[wv];
        s += b[0];
        out[g] = 1.0f / (1.0f + expf(-s));
    }
}

// ---------------------------------------------------------------------------
extern "C" void kernel_launch(void* const* d_in, const int* in_sizes, int n_in,
                              void* d_out, int out_size, void* d_ws, size_t ws_size,
                              hipStream_t stream) {
    const float* X    = (const float*)d_in[0];
    const int*   ei   = (const int*)  d_in[1];   // [2, E] row-major: src row then dst row
    const int*   batc = (const int*)  d_in[2];
    const float* W1   = (const float*)d_in[3];
    const float* b1   = (const float*)d_in[4];
    const float* W2   = (const float*)d_in[5];
    const float* b2   = (const float*)d_in[6];
    const float* Wm1  = (const float*)d_in[7];
    const float* bm1  = (const float*)d_in[8];
    const float* Wm2  = (const float*)d_in[9];
    const float* bm2  = (const float*)d_in[10];
    float* out = (float*)d_out;

    const int* src = ei;
    const int* dst = ei + NE;

    // workspace carve-out
    char* ws = (char*)d_ws;
    size_t off = 0;
    auto carve = [&](size_t bytes) { void* p = ws + off; off += (bytes + 255) & ~(size_t)255; return p; };
    float* bufA = (float*)carve((size_t)NN * DH * sizeof(float));   // gemm outputs
    float* bufB = (float*)carve((size_t)NN * DH * sizeof(float));   // aggregation
    float* isq  = (float*)carve((size_t)NN * sizeof(float));        // deg -> deg^-1/2
    float* pool = (float*)carve((size_t)NG * DH * sizeof(float));
    float* cnt  = (float*)carve((size_t)NG * sizeof(float));
    float* zbuf = (float*)carve((size_t)NG * DMLP * sizeof(float));
    (void)ws_size; (void)in_sizes; (void)n_in; (void)out_size;

    // 1) degree / normalization
    hipMemsetAsync(isq, 0, (size_t)NN * sizeof(float), stream);
    deg_count_kernel<<<(NE + 255) / 256, 256, 0, stream>>>(dst, isq, NE);
    deg_finalize_kernel<<<(NN + 255) / 256, 256, 0, stream>>>(isq, NN);

    // 2) GEMM1: bufA = X @ W1    [100000x128]@[128x256]
    {
        const int tiles = (NN / 32) * (DH / 64);            // 12500 wave-tiles
        wmma_gemm_f32_kernel<DIN, DH, false, false>
            <<<(tiles + 7) / 8, 256, 0, stream>>>(X, W1, nullptr, bufA, NN);
    }

    // 3) scatter conv1 -> bufB; then bias+relu in place
    hipMemsetAsync(bufB, 0, (size_t)NN * DH * sizeof(float), stream);
    gcn_scatter_kernel<<<NE + NN, 64, 0, stream>>>(src, dst, isq, bufA, bufB, NE);
    bias_relu_kernel<<<(NN * DH + 255) / 256, 256, 0, stream>>>(bufB, b1, NN * DH, DH - 1);

    // 4) GEMM2: bufA = bufB @ W2   [100000x256]@[256x256]
    {
        const int tiles = (NN / 32) * (DH / 64);
        wmma_gemm_f32_kernel<DH, DH, false, false>
            <<<(tiles + 7) / 8, 256, 0, stream>>>(bufB, W2, nullptr, bufA, NN);
    }

    // 5) scatter conv2 -> bufB  (b2 folded into pool finalize)
    hipMemsetAsync(bufB, 0, (size_t)NN * DH * sizeof(float), stream);
    gcn_scatter_kernel<<<NE + NN, 64, 0, stream>>>(src, dst, isq, bufA, bufB, NE);

    // 6) mean pool
    hipMemsetAsync(pool, 0, (size_t)NG * DH * sizeof(float), stream);
    hipMemsetAsync(cnt, 0, (size_t)NG * sizeof(float), stream);
    pool_accum_kernel<<<NN, 256, 0, stream>>>(bufB, batc, pool, cnt);
    pool_final_kernel<<<(NG * DH + 255) / 256, 256, 0, stream>>>(pool, cnt, b2, NG * DH, DH - 1, 8);

    // 7) MLP layer 1 with fused bias+relu: zbuf = relu(pool @ Wm1 + bm1)  [512x256]@[256x512]
    {
        const int tiles = (NG / 32) * (DMLP / 64);          // 128 wave-tiles -> 16 blocks
        wmma_gemm_f32_kernel<DH, DMLP, true, true>
            <<<(tiles + 7) / 8, 256, 0, stream>>>(pool, Wm1, bm1, zbuf, NG);
    }

    // 8) head: out = sigmoid(zbuf @ Wm2 + bm2)
    head_kernel<<<NG, 256, 0, stream>>>(zbuf, Wm2, bm2, out, DMLP);
}